// GraphSAGE_13589276525101
// MI455X (gfx1250) — compile-verified
//
#include <hip/hip_runtime.h>

typedef float v2f __attribute__((ext_vector_type(2)));
typedef float v4f __attribute__((ext_vector_type(4)));
typedef float v8f __attribute__((ext_vector_type(8)));

#define DH 128

// ---------------------------------------------------------------------------
// Zero a float buffer
// ---------------------------------------------------------------------------
__global__ void sage_zero(float* __restrict__ p, long long n) {
    long long i = (long long)blockIdx.x * blockDim.x + threadIdx.x;
    if (i < n) p[i] = 0.0f;
}

// ---------------------------------------------------------------------------
// Fused edge scatter: agg[dst[e]] += h[src[e]]  (128 floats per edge)
// Each thread handles 4 consecutive floats (one b128 load + 4 f32 atomics);
// one wave covers one edge row (32 lanes x 4 = 128 floats), fully coalesced.
// Lane 0 of each edge also bumps deg[dst[e]] when add_deg is set.
// ---------------------------------------------------------------------------
__global__ void sage_scatter(const float* __restrict__ h,
                             const int* __restrict__ src,
                             const int* __restrict__ dst,
                             float* __restrict__ agg,
                             float* __restrict__ deg,
                             long long total, int add_deg) {
    long long idx = (long long)blockIdx.x * blockDim.x + threadIdx.x;
    if (idx >= total) return;                 // total = E * 32
    long long e = idx >> 5;
    int f4 = (int)(idx & 31) << 2;            // 0,4,...,124
    int s = src[e];
    int d = dst[e];
    v4f v = *(const v4f*)(h + (long long)s * DH + f4);
    float* ap = agg + (long long)d * DH + f4;
    atomicAdd(ap + 0, v.x);
    atomicAdd(ap + 1, v.y);
    atomicAdd(ap + 2, v.z);
    atomicAdd(ap + 3, v.w);
    if (add_deg && (idx & 31) == 0) atomicAdd(&deg[d], 1.0f);
}

// ---------------------------------------------------------------------------
// WMMA GEMM:  out = act( A_self @ Wself^T + (agg/deg) @ Wneigh^T + bias )
//
// Block = 256 threads (8 waves); wave w owns 16-row strip (blockIdx*8 + w).
// Both weight matrices are staged in LDS (2 x 64 KB) once per block; the
// K-loop reads B fragments with ds_load_b64 and batches all 16 fragments
// into distinct registers before issuing 16 back-to-back
// V_WMMA_F32_16X16X4_F32 ops (full fp32 precision, K=4 per step, 32 steps).
//
// fp32 fragment layouts (ISA 7.12.2):
//   A 16x4 : lane L -> M = L&15, VGPR0/1 = K = k0 + 2*(L>>4) (+1)
//   B 4x16 : lane L -> N = L&15, VGPR0/1 = K = k0 + 2*(L>>4) (+1)
//   C/D    : lane L -> N = L&15, VGPR r  -> M = r + 8*(L>>4)
// ---------------------------------------------------------------------------
__global__ __launch_bounds__(256)
void sage_gemm(const float* __restrict__ Aself,
               const float* __restrict__ agg,
               const float* __restrict__ deg,
               const float* __restrict__ Wself,
               const float* __restrict__ Wneigh,
               const float* __restrict__ bias,
               float* __restrict__ out,
               int N, int do_relu) {
    __shared__ float ldsWs[DH * DH];   // 64 KB
    __shared__ float ldsWn[DH * DH];   // 64 KB

    // ---- cooperative weight staging (b128 loads) ----
    {
        const v4f* gs = (const v4f*)Wself;
        const v4f* gn = (const v4f*)Wneigh;
        v4f* ls = (v4f*)ldsWs;
        v4f* ln = (v4f*)ldsWn;
        for (int i = threadIdx.x; i < DH * DH / 4; i += 256) {
            ls[i] = gs[i];
            ln[i] = gn[i];
        }
    }
    __syncthreads();

    const int lane = threadIdx.x & 31;
    const int wave = threadIdx.x >> 5;
    const int half = lane >> 4;
    const int m16  = lane & 15;

    const int nStrips = (N + 15) / 16;
    int strip = blockIdx.x * 8 + wave;
    if (strip >= nStrips) strip = nStrips - 1;   // duplicate work; identical stores
    const int row0 = strip * 16;

    int rowA = row0 + m16;
    if (rowA > N - 1) rowA = N - 1;

    float dv = deg[rowA];
    float invd = 1.0f / fmaxf(dv, 1.0f);

    const long long aBase = (long long)rowA * DH + 2 * half;
    const int bOff = m16 * DH + 2 * half;        // LDS fragment base (c adds c*16*DH)

    v8f acc[8];
#pragma unroll
    for (int c = 0; c < 8; ++c) acc[c] = (v8f)0.0f;

#pragma unroll 2
    for (int k0 = 0; k0 < DH; k0 += 4) {
        v2f a_s = *(const v2f*)(Aself + aBase + k0);
        v2f a_n = *(const v2f*)(agg   + aBase + k0);
        a_n.x *= invd;
        a_n.y *= invd;

        v2f bs[8], bn[8];
#pragma unroll
        for (int c = 0; c < 8; ++c) {
            bs[c] = *(const v2f*)(&ldsWs[c * 16 * DH + bOff + k0]);
            bn[c] = *(const v2f*)(&ldsWn[c * 16 * DH + bOff + k0]);
        }
#pragma unroll
        for (int c = 0; c < 8; ++c) {
            acc[c] = __builtin_amdgcn_wmma_f32_16x16x4_f32(
                false, a_s, false, bs[c], (short)0, acc[c], false, false);
            acc[c] = __builtin_amdgcn_wmma_f32_16x16x4_f32(
                false, a_n, false, bn[c], (short)0, acc[c], false, false);
        }
    }

    // ---- epilogue: bias, optional ReLU, store ----
#pragma unroll
    for (int c = 0; c < 8; ++c) {
        float bv = bias[c * 16 + m16];
#pragma unroll
        for (int r = 0; r < 8; ++r) {
            int rowOut = row0 + r + 8 * half;
            float v = acc[c][r] + bv;
            if (do_relu) v = fmaxf(v, 0.0f);
            if (rowOut < N) out[(long long)rowOut * DH + c * 16 + m16] = v;
        }
    }
}

// ---------------------------------------------------------------------------
// Host-side orchestration
// inputs: x[N*128], src[E], dst[E], Ws1, Wn1, b1, Ws2, Wn2, b2
// ws: deg (padded N) | agg (N*128).  d_out doubles as the h1 buffer (each
// GEMM wave reads/writes only its own strip; kernels are stream-ordered).
// ---------------------------------------------------------------------------
extern "C" void kernel_launch(void* const* d_in, const int* in_sizes, int n_in,
                              void* d_out, int out_size, void* d_ws, size_t ws_size,
                              hipStream_t stream) {
    const float* x   = (const float*)d_in[0];
    const int*   src = (const int*)  d_in[1];
    const int*   dst = (const int*)  d_in[2];
    const float* Ws1 = (const float*)d_in[3];
    const float* Wn1 = (const float*)d_in[4];
    const float* b1  = (const float*)d_in[5];
    const float* Ws2 = (const float*)d_in[6];
    const float* Wn2 = (const float*)d_in[7];
    const float* b2  = (const float*)d_in[8];
    float* out = (float*)d_out;

    const int N = in_sizes[0] / DH;
    const int E = in_sizes[1];

    const long long padN = ((long long)N + 63) & ~63LL;
    float* deg = (float*)d_ws;
    float* agg = deg + padN;

    const long long featTotal = (long long)N * DH;
    const long long zeroTotal = padN + featTotal;
    const long long edgeTotal = (long long)E * 32;   // 4 floats per thread

    const int ZB = 256;
    const int SB = 256;
    const int gemmBlocks = ((N + 15) / 16 + 7) / 8;

    // ---- layer 1 ----
    sage_zero<<<(unsigned)((zeroTotal + ZB - 1) / ZB), ZB, 0, stream>>>(deg, zeroTotal);
    sage_scatter<<<(unsigned)((edgeTotal + SB - 1) / SB), SB, 0, stream>>>(
        x, src, dst, agg, deg, edgeTotal, /*add_deg=*/1);
    sage_gemm<<<gemmBlocks, 256, 0, stream>>>(
        x, agg, deg, Ws1, Wn1, b1, out, N, /*relu=*/1);

    // ---- layer 2 ----
    sage_zero<<<(unsigned)((featTotal + ZB - 1) / ZB), ZB, 0, stream>>>(agg, featTotal);
    sage_scatter<<<(unsigned)((edgeTotal + SB - 1) / SB), SB, 0, stream>>>(
        out, src, dst, agg, deg, edgeTotal, /*add_deg=*/0);
    sage_gemm<<<gemmBlocks, 256, 0, stream>>>(
        out, agg, deg, Ws2, Wn2, b2, out, N, /*relu=*/0);
}